// HeatmapOffsetmapLoss_41412074668387
// MI455X (gfx1250) — compile-verified
//
#include <hip/hip_runtime.h>
#include <math.h>

#define HH 512
#define WW 512
#define NPTS 19
#define BATCH 4
#define RAD2 1600
#define NBLOCKS 1024
#define NTHREADS 256

typedef __attribute__((ext_vector_type(2))) float v2f;
typedef __attribute__((ext_vector_type(8))) float v8f;

__device__ __forceinline__ float wave_reduce_full(float v) {
#pragma unroll
    for (int m = 16; m >= 1; m >>= 1) v += __shfl_xor(v, m, 32);
    return v;
}

// Pass 1: grid-stride streaming reduction over feature_maps (float4 loads).
// Emits per-block partials (bce_sum, l1_sum, cnt) into d_ws, SoA layout.
__global__ __launch_bounds__(NTHREADS) void hm_loss_partial(
    const float* __restrict__ fm, const int* __restrict__ lm,
    float* __restrict__ partial)
{
    const int total4 = BATCH * NPTS * HH * (WW / 4);  // 4,980,736 float4 units
    float s_bce = 0.f, s_l1 = 0.f, s_cnt = 0.f;

    for (int u = blockIdx.x * blockDim.x + threadIdx.x; u < total4;
         u += gridDim.x * blockDim.x) {
        int j4 = u & 127;          // WW/4 = 128
        int t  = u >> 7;
        int i  = t & 511;          // HH = 512
        t >>= 9;
        int n  = t % NPTS;
        int b  = t / NPTS;

        int x = lm[(b * NPTS + n) * 2 + 0];
        int y = lm[(b * NPTS + n) * 2 + 1];
        x = min(max(x, 1), WW - 1);
        y = min(max(y, 1), HH - 1);

        int j = j4 << 2;
        size_t planeoff = (size_t)i * WW + j;
        const float4 ph4 = *(const float4*)(fm + (size_t)(b * 3 * NPTS + n)            * (HH * WW) + planeoff);
        const float4 px4 = *(const float4*)(fm + (size_t)(b * 3 * NPTS + NPTS + n)     * (HH * WW) + planeoff);
        const float4 py4 = *(const float4*)(fm + (size_t)(b * 3 * NPTS + 2 * NPTS + n) * (HH * WW) + planeoff);

        int dy  = i - y;
        int dy2 = dy * dy;
        float tx = (float)(y - i) / 40.0f;   // cropped offsetmap_x value

        float phv[4] = {ph4.x, ph4.y, ph4.z, ph4.w};
        float pxv[4] = {px4.x, px4.y, px4.z, px4.w};
        float pyv[4] = {py4.x, py4.y, py4.z, py4.w};

#pragma unroll
        for (int k = 0; k < 4; ++k) {
            int dx = j + k - x;
            bool inside = (dy2 + dx * dx) <= RAD2;   // cropped heatmap == 1
            float ph = phv[k];
            // stable BCE-with-logits term: max(ph,0) - ph*th + log1p(exp(-|ph|))
            float sp = __logf(1.0f + __expf(-fabsf(ph)));
            s_bce += fmaxf(ph, 0.f) + sp - (inside ? ph : 0.f);
            float ty  = (float)(-dx) / 40.0f;        // (x - (j+k)) / 40
            float l1t = fabsf(pxv[k] - tx) + fabsf(pyv[k] - ty);
            s_l1  += inside ? l1t : 0.f;
            s_cnt += inside ? 1.f : 0.f;
        }
    }

    s_bce = wave_reduce_full(s_bce);
    s_l1  = wave_reduce_full(s_l1);
    s_cnt = wave_reduce_full(s_cnt);

    __shared__ float red[3][NTHREADS / 32];
    int wave = threadIdx.x >> 5, lane = threadIdx.x & 31;
    if (lane == 0) { red[0][wave] = s_bce; red[1][wave] = s_l1; red[2][wave] = s_cnt; }
    __syncthreads();
    if (threadIdx.x == 0) {
        float t0 = 0.f, t1 = 0.f, t2 = 0.f;
#pragma unroll
        for (int w = 0; w < NTHREADS / 32; ++w) { t0 += red[0][w]; t1 += red[1][w]; t2 += red[2][w]; }
        partial[blockIdx.x]               = t0;
        partial[NBLOCKS + blockIdx.x]     = t1;
        partial[2 * NBLOCKS + blockIdx.x] = t2;
    }
}

// Pass 2: one wave32. Per-lane strided sums, 4x shfl_xor stages to reduce
// within each 16-lane half, then V_WMMA_F32_16X16X4_F32 (A = ones, B.vgpr0 =
// per-lane partials, B.vgpr1 = 0) so D[m,n] = B[n] + B[n+16] = grand total in
// every D element. EXEC is all-ones (full wave, no divergence) as WMMA requires.
__global__ __launch_bounds__(32) void hm_loss_final(
    const float* __restrict__ partial, float* __restrict__ out)
{
    int lane = threadIdx.x;
    float s0 = 0.f, s1 = 0.f, s2 = 0.f;
    for (int p = lane; p < NBLOCKS; p += 32) {
        s0 += partial[p];
        s1 += partial[NBLOCKS + p];
        s2 += partial[2 * NBLOCKS + p];
    }
#pragma unroll
    for (int m = 1; m <= 8; m <<= 1) {
        s0 += __shfl_xor(s0, m, 32);
        s1 += __shfl_xor(s1, m, 32);
        s2 += __shfl_xor(s2, m, 32);
    }
    // Now lanes 0..15 hold the low-half sum, lanes 16..31 the high-half sum.
    v2f a; a.x = 1.0f; a.y = 1.0f;   // A = ones (16x4)
    v8f c = {};
    v2f b0; b0.x = s0; b0.y = 0.0f;
    v8f d0 = __builtin_amdgcn_wmma_f32_16x16x4_f32(false, a, false, b0, (short)0, c, false, false);
    v2f b1; b1.x = s1; b1.y = 0.0f;
    v8f d1 = __builtin_amdgcn_wmma_f32_16x16x4_f32(false, a, false, b1, (short)0, c, false, false);
    v2f b2; b2.x = s2; b2.y = 0.0f;
    v8f d2 = __builtin_amdgcn_wmma_f32_16x16x4_f32(false, a, false, b2, (short)0, c, false, false);

    if (lane == 0) {
        float tot_bce = d0[0];
        float tot_l1  = d1[0];
        float tot_cnt = d2[0];
        const float denom = (float)(BATCH * NPTS) * (float)(HH * WW); // 19,922,944
        out[0] = 2.0f * (tot_bce / denom) + tot_l1 / tot_cnt;
    }
}

extern "C" void kernel_launch(void* const* d_in, const int* in_sizes, int n_in,
                              void* d_out, int out_size, void* d_ws, size_t ws_size,
                              hipStream_t stream) {
    const float* fm = (const float*)d_in[0];   // [4, 57, 512, 512] f32
    const int*   lm = (const int*)d_in[1];     // [4, 19, 2] i32
    float* part = (float*)d_ws;                // 3 * NBLOCKS floats = 12 KB
    hm_loss_partial<<<NBLOCKS, NTHREADS, 0, stream>>>(fm, lm, part);
    hm_loss_final<<<1, 32, 0, stream>>>(part, (float*)d_out);
}